// KernelInduced_30494267802193
// MI455X (gfx1250) — compile-verified
//
#include <hip/hip_runtime.h>

// ---------------------------------------------------------------------------
// Multipole graph kernel network (MGKN) for MI455X / gfx1250.
// Per-edge kernel-MLP final layer fused with the message contraction via
// V_WMMA_F32_16X16X4_F32 (fp32, matches reference numerics).
// Final-layer weights are staged per 32-column chunk into LDS with the
// Tensor Data Mover (double-buffered, TENSORcnt-tracked) and shared by
// 8 waves (128 edges) per workgroup.
// ---------------------------------------------------------------------------

#define NN   10752     // total nodes
#define NP0  8192      // level-0 nodes
#define WID  32        // feature width
#define NSEG 7         // dn0, dn1, mid0, mid1, mid2, up0, up1
#define WAVES 8        // waves per block
#define EPB  (WAVES * 16)  // edges per block

typedef float v2f __attribute__((ext_vector_type(2)));
typedef float v8f __attribute__((ext_vector_type(8)));

#if __has_builtin(__builtin_amdgcn_tensor_load_to_lds) && \
    __has_builtin(__builtin_amdgcn_s_wait_tensorcnt)
#define USE_TDM 1
#else
#define USE_TDM 0
#endif

#if USE_TDM
typedef unsigned int tdm_v4u __attribute__((ext_vector_type(4)));
typedef int          tdm_v8i __attribute__((ext_vector_type(8)));
typedef int          tdm_v4i __attribute__((ext_vector_type(4)));

// DMA a K x 32-DWORD tile (row stride 1024 floats in memory) into LDS with
// 8-DWORD padding after each 32-DWORD row => LDS row stride 40 floats.
// D# layout per CDNA5 ISA 8.3/8.4 (group0 128b, group1 256b).
template<int K>
__device__ __forceinline__ void tdm_load_chunk(const float* gsrc, unsigned lds_off) {
  unsigned long long ga = (unsigned long long)(size_t)gsrc;
  tdm_v4u g0;
  g0[0] = 1u;                                    // count=1, user mode
  g0[1] = lds_off;                               // lds_addr (bytes)
  g0[2] = (unsigned)(ga & 0xFFFFFFFFu);          // global_addr[31:0]
  g0[3] = (unsigned)((ga >> 32) & 0x01FFFFFFu)   // global_addr[56:32]
        | (2u << 30);                            // type=2 ("image")
  tdm_v8i g1;
  g1[0] = (int)((2u << 16)      // data_size = 4B
        |       (1u << 20)      // pad_enable
        |       (4u << 22)      // pad_interval: 32 DWORDs
        |       (7u << 25));    // pad_amount: 8 DWORDs
  g1[1] = (int)((1024u & 0xFFFFu) << 16);        // tensor_dim0[15:0] = 1024
  g1[2] = (int)(((unsigned)K & 0xFFFFu) << 16);  // tensor_dim1[15:0] = K
  g1[3] = (int)(32u << 16);                      // tile_dim0 = 32
  g1[4] = (int)K;                                // tile_dim1 = K
  g1[5] = 1024;                                  // tensor_dim0_stride
  g1[6] = 0;
  g1[7] = 0;
  tdm_v4i gz = {0, 0, 0, 0};
#if __clang_major__ >= 23
  tdm_v8i gz8 = {0, 0, 0, 0, 0, 0, 0, 0};
  __builtin_amdgcn_tensor_load_to_lds(g0, g1, gz, gz, gz8, 0);
#else
  __builtin_amdgcn_tensor_load_to_lds(g0, g1, gz, gz, 0);
#endif
}
#endif // USE_TDM

__device__ __forceinline__ v8f wmma4(v2f a, v2f b, v8f c) {
  // D = A(16x4,f32) * B(4x16,f32) + C(16x16,f32)
  return __builtin_amdgcn_wmma_f32_16x16x4_f32(
      /*neg_a=*/false, a, /*neg_b=*/false, b,
      /*c_mod=*/(short)0, c, /*reuse_a=*/false, /*reuse_b=*/false);
}

// 8 waves per block; each wave owns a tile of 16 edges. All edge segments are
// multiples of 128 (DEG==16), so tiles are always full and EXEC stays all-ones
// around every WMMA.
template<int K, bool THREE>
__global__ __launch_bounds__(256)
void nnconv_edge_kernel(const int* __restrict__ eidx, int Etot,
                        const float* __restrict__ ea, int base,
                        const float* __restrict__ hfeat, float* __restrict__ acc,
                        const float* __restrict__ W1, const float* __restrict__ b1,
                        const float* __restrict__ W2, const float* __restrict__ b2,
                        const float* __restrict__ WF, const float* __restrict__ bF)
{
  constexpr int KS = K / 4;           // K-steps per 16x16 WMMA tile
  constexpr int CH = K * 40;          // staged chunk: K rows, stride 40 floats

  __shared__ float sEA[WAVES][16 * 8];            // edge attrs (padded 6->8)
  __shared__ float sH [WAVES][16 * 68];           // hidden layer 1
  __shared__ float sH2[THREE ? WAVES : 1][16 * 68]; // hidden layer 2
  __shared__ float sX [WAVES][16 * 32];           // x[src] feature rows
  __shared__ int   sSRC[WAVES][16];
  __shared__ int   sDST[WAVES][16];
  __shared__ float sWF[2][CH];                    // double-buffered W_F chunk
  __shared__ float sBF[1024];                     // final bias

  const int tid   = (int)threadIdx.x;
  const int wv    = tid >> 5;          // wave in block
  const int ln    = tid & 31;          // lane
  const int ncol  = ln & 15;           // N column of WMMA fragments
  const int hi    = ln >> 4;           // lane half
  const int kofs  = hi * 2;            // K offset within a 4-wide K step
  const int row0  = hi * 8;            // first C/D row held by this lane

#if USE_TDM
  // Kick off chunk 0 while the per-wave setup runs.
  if (wv == 0)
    tdm_load_chunk<K>(WF, (unsigned)(size_t)&sWF[0][0]);
#endif

  // Stage final bias once per block (4 KB).
  ((float4*)sBF)[tid] = ((const float4*)bF)[tid];

  const int e0 = base + ((int)blockIdx.x * WAVES + wv) * 16;
  if (ln < 16) {
    const int e = e0 + ln;
    sSRC[wv][ln] = eidx[e];
    sDST[wv][ln] = eidx[Etot + e];
#pragma unroll
    for (int c = 0; c < 6; ++c) sEA[wv][ln * 8 + c] = ea[(long)e * 6 + c];
  }
  __syncthreads();

  // ---- layer 1 (6 -> K), tiny K-dim: plain VALU --------------------------
  for (int idx = ln; idx < 16 * K; idx += 32) {
    const int m = idx / K, k = idx - m * K;
    float a = b1[k];
#pragma unroll
    for (int c = 0; c < 6; ++c) a += sEA[wv][m * 8 + c] * W1[c * K + k];
    sH[wv][m * 68 + k] = fmaxf(a, 0.f);
  }
  __syncthreads();

  // ---- layer 2 (K -> K) for 3-layer MLPs: WMMA ---------------------------
  if (THREE) {
    v2f af[KS];
#pragma unroll
    for (int ks = 0; ks < KS; ++ks) {
      af[ks].x = sH[wv][ncol * 68 + ks * 4 + kofs];
      af[ks].y = sH[wv][ncol * 68 + ks * 4 + kofs + 1];
    }
#pragma unroll
    for (int nt = 0; nt < K / 16; ++nt) {
      const float bias = b2[nt * 16 + ncol];
      v8f c = {bias, bias, bias, bias, bias, bias, bias, bias};
#pragma unroll
      for (int ks = 0; ks < KS; ++ks) {
        v2f b;
        b.x = W2[(ks * 4 + kofs)     * K + nt * 16 + ncol];
        b.y = W2[(ks * 4 + kofs + 1) * K + nt * 16 + ncol];
        c = wmma4(af[ks], b, c);
      }
#pragma unroll
      for (int v = 0; v < 8; ++v)
        sH2[wv][(row0 + v) * 68 + nt * 16 + ncol] = fmaxf(c[v], 0.f);
    }
    __syncthreads();
  }
  const float* hsrc = THREE ? sH2[wv] : sH[wv];

  // ---- gather x[src] rows ------------------------------------------------
#pragma unroll
  for (int q = 0; q < 16; ++q) {
    const int idx = ln + q * 32;
    const int m = idx >> 5, i = idx & 31;
    sX[wv][m * 32 + i] = hfeat[(long)sSRC[wv][m] * WID + i];
  }

#if !USE_TDM
  // Fallback staging of chunk 0.
  for (int t = tid; t < K * 32; t += 256) {
    const int r = t >> 5, c = t & 31;
    sWF[0][r * 40 + c] = WF[(long)r * 1024 + c];
  }
#endif
  __syncthreads();

  // A-fragments of the final layer: identical for every output column chunk.
  v2f afr[KS];
#pragma unroll
  for (int ks = 0; ks < KS; ++ks) {
    afr[ks].x = hsrc[ncol * 68 + ks * 4 + kofs];
    afr[ks].y = hsrc[ncol * 68 + ks * 4 + kofs + 1];
  }

  // ---- fused final layer (K -> 1024) + message contraction ---------------
  // msg[e, j] = sum_i x[src_e, i] * (hlast[e,:] @ WF[:, i*32+j] + bF[i*32+j])
  v8f msgA = {0, 0, 0, 0, 0, 0, 0, 0};
  v8f msgB = {0, 0, 0, 0, 0, 0, 0, 0};
#pragma unroll 1
  for (int i = 0; i < 32; ++i) {
    // Stage chunk i+1 into the other buffer while computing with chunk i.
#if USE_TDM
    if (wv == 0) {
      if (i + 1 < 32) {
        tdm_load_chunk<K>(WF + (long)(i + 1) * 32,
                          (unsigned)(size_t)&sWF[(i + 1) & 1][0]);
        __builtin_amdgcn_s_wait_tensorcnt(1);   // chunk i complete
      } else {
        __builtin_amdgcn_s_wait_tensorcnt(0);   // last chunk complete
      }
    }
#else
    if (i + 1 < 32) {
      float* nxt = sWF[(i + 1) & 1];
      for (int t = tid; t < K * 32; t += 256) {
        const int r = t >> 5, c = t & 31;
        nxt[r * 40 + c] = WF[(long)r * 1024 + (i + 1) * 32 + c];
      }
    }
#endif
    __syncthreads();    // chunk i visible to all waves

    const float* cur = sWF[i & 1];
    float xr[8];
#pragma unroll
    for (int v = 0; v < 8; ++v) xr[v] = sX[wv][(row0 + v) * 32 + i];
#pragma unroll
    for (int nt = 0; nt < 2; ++nt) {
      const int col2 = nt * 16 + ncol;
      const float bias = sBF[i * 32 + col2];
      v8f c = {bias, bias, bias, bias, bias, bias, bias, bias};
#pragma unroll
      for (int ks = 0; ks < KS; ++ks) {
        v2f b;
        b.x = cur[(ks * 4 + kofs)     * 40 + col2];
        b.y = cur[(ks * 4 + kofs + 1) * 40 + col2];
        c = wmma4(afr[ks], b, c);
      }
      if (nt == 0) {
#pragma unroll
        for (int v = 0; v < 8; ++v) msgA[v] += xr[v] * c[v];
      } else {
#pragma unroll
        for (int v = 0; v < 8; ++v) msgB[v] += xr[v] * c[v];
      }
    }
    __syncthreads();    // all waves done with chunk i before it is overwritten
  }

  // ---- scatter-add into per-node accumulator -----------------------------
#pragma unroll
  for (int v = 0; v < 8; ++v) {
    const int d = sDST[wv][row0 + v];
    atomicAdd(&acc[(long)d * WID + ncol],      msgA[v]);
    atomicAdd(&acc[(long)d * WID + 16 + ncol], msgB[v]);
  }
}

// ---------------------------------------------------------------------------

__global__ void zero_u32_kernel(unsigned* __restrict__ p, int n) {
  int i = blockIdx.x * blockDim.x + threadIdx.x;
  if (i < n) p[i] = 0u;
}

__global__ void count_kernel(const int* __restrict__ eidx, int Etot, int base,
                             int Eseg, int* __restrict__ cnt) {
  int t = blockIdx.x * blockDim.x + threadIdx.x;
  if (t < Eseg) atomicAdd(&cnt[eidx[Etot + base + t]], 1);
}

__global__ void inv_kernel(const int* __restrict__ cnt, float* __restrict__ invc, int n) {
  int i = blockIdx.x * blockDim.x + threadIdx.x;
  if (i < n) invc[i] = 1.f / fmaxf((float)cnt[i], 1.f);
}

__global__ void fc_in_kernel(const float* __restrict__ x, const float* __restrict__ W,
                             const float* __restrict__ b, float* __restrict__ h) {
  int idx = blockIdx.x * blockDim.x + threadIdx.x;
  if (idx < NN * WID) {
    int node = idx >> 5, j = idx & 31;
    h[idx] = x[node] * W[j] + b[j];
  }
}

// h = relu(h + acc * inv_cnt); reset acc for the next conv.
__global__ void node_update_kernel(float* __restrict__ h, float* __restrict__ acc,
                                   const float* __restrict__ invc_seg) {
  int idx = blockIdx.x * blockDim.x + threadIdx.x;
  if (idx < NN * WID) {
    float inv = invc_seg[idx >> 5];
    h[idx] = fmaxf(h[idx] + acc[idx] * inv, 0.f);
    acc[idx] = 0.f;
  }
}

__global__ void fc_out1_kernel(const float* __restrict__ h, const float* __restrict__ W,
                               const float* __restrict__ b, float* __restrict__ h2) {
  int idx = blockIdx.x * blockDim.x + threadIdx.x;
  if (idx < NP0 * 64) {
    int n = idx >> 6, o = idx & 63;
    float a = b[o];
#pragma unroll 8
    for (int i = 0; i < 32; ++i) a += h[n * 32 + i] * W[i * 64 + o];
    h2[idx] = fmaxf(a, 0.f);
  }
}

__global__ void fc_out2_kernel(const float* __restrict__ h2, const float* __restrict__ W,
                               const float* __restrict__ b, float* __restrict__ out) {
  int n = blockIdx.x * blockDim.x + threadIdx.x;
  if (n < NP0) {
    float a = b[0];
#pragma unroll 8
    for (int o = 0; o < 64; ++o) a += h2[n * 64 + o] * W[o];
    out[n] = a;
  }
}

// ---------------------------------------------------------------------------

struct ConvDesc {
  const int* ei; int Etot;
  const float* ea; int base; int Eseg;
  const float *W1, *b1, *W2, *b2, *WF, *bF;
  int K; int three; int seg;
};

extern "C" void kernel_launch(void* const* d_in, const int* in_sizes, int n_in,
                              void* d_out, int out_size, void* d_ws, size_t ws_size,
                              hipStream_t stream) {
  (void)in_sizes; (void)n_in; (void)out_size; (void)ws_size;

  const float* x      = (const float*)d_in[0];
  const int*   ei_dn  = (const int*)  d_in[1];
  const float* ea_dn  = (const float*)d_in[2];
  const int*   ei_mid = (const int*)  d_in[4];
  const float* ea_mid = (const float*)d_in[5];
  const int*   ei_up  = (const int*)  d_in[7];
  const float* ea_up  = (const float*)d_in[8];
  const float* fcin_W = (const float*)d_in[10];
  const float* fcin_b = (const float*)d_in[11];
  // down_kernels: l=1 (K=32): 12..15 ; l=2 (K=16): 16..19  (W1,b1,WF,bF)
  // mid_kernels:  l=0 (K=64): 20..25 ; l=1 (K=32): 26..31 ; l=2 (K=16): 32..37
  // up_kernels:   l=1 (K=32): 38..41 ; l=2 (K=16): 42..45
  const float* fo1_W = (const float*)d_in[46];
  const float* fo1_b = (const float*)d_in[47];
  const float* fo2_W = (const float*)d_in[48];
  const float* fo2_b = (const float*)d_in[49];

  const int E_dn = 40960, E_mid = 172032, E_up = 40960;

  auto F = [&](int i) { return (const float*)d_in[i]; };

  // Per-depth conv order: down0, down1, mid2, up1, mid1, up0, mid0.
  ConvDesc convs[NSEG] = {
    { ei_dn,  E_dn,  ea_dn,       0,  32768, F(12), F(13), nullptr, nullptr, F(14), F(15), 32, 0, 0 },
    { ei_dn,  E_dn,  ea_dn,   32768,   8192, F(16), F(17), nullptr, nullptr, F(18), F(19), 16, 0, 1 },
    { ei_mid, E_mid, ea_mid, 163840,   8192, F(32), F(33), F(34),   F(35),   F(36), F(37), 16, 1, 4 },
    { ei_up,  E_up,  ea_up,   32768,   8192, F(42), F(43), nullptr, nullptr, F(44), F(45), 16, 0, 6 },
    { ei_mid, E_mid, ea_mid, 131072,  32768, F(26), F(27), F(28),   F(29),   F(30), F(31), 32, 1, 3 },
    { ei_up,  E_up,  ea_up,       0,  32768, F(38), F(39), nullptr, nullptr, F(40), F(41), 32, 0, 5 },
    { ei_mid, E_mid, ea_mid,      0, 131072, F(20), F(21), F(22),   F(23),   F(24), F(25), 64, 1, 2 },
  };

  // Workspace layout (floats).
  float* h    = (float*)d_ws;        // NN*32
  float* acc  = h    + NN * WID;     // NN*32
  float* h2   = acc  + NN * WID;     // NP0*64
  float* invc = h2   + NP0 * 64;     // NSEG*NN
  int*   cnt  = (int*)(invc + NSEG * NN); // NSEG*NN

  const int TB = 256;
  zero_u32_kernel<<<(NN * WID + TB - 1) / TB, TB, 0, stream>>>((unsigned*)acc, NN * WID);
  zero_u32_kernel<<<(NSEG * NN + TB - 1) / TB, TB, 0, stream>>>((unsigned*)cnt, NSEG * NN);

  // In-degree counts per segment (static across depth iterations).
  for (int s = 0; s < NSEG; ++s) {
    const ConvDesc& c = convs[s];
    count_kernel<<<(c.Eseg + TB - 1) / TB, TB, 0, stream>>>(
        c.ei, c.Etot, c.base, c.Eseg, cnt + (long)c.seg * NN);
  }
  inv_kernel<<<(NSEG * NN + TB - 1) / TB, TB, 0, stream>>>(cnt, invc, NSEG * NN);

  fc_in_kernel<<<(NN * WID + TB - 1) / TB, TB, 0, stream>>>(x, fcin_W, fcin_b, h);

  for (int depth = 0; depth < 4; ++depth) {
    for (int s = 0; s < NSEG; ++s) {
      const ConvDesc& c = convs[s];
      dim3 grid(c.Eseg / EPB), blk(256);
#define LAUNCH_CONV(KV, TV)                                                    \
      nnconv_edge_kernel<KV, TV><<<grid, blk, 0, stream>>>(                    \
          c.ei, c.Etot, c.ea, c.base, h, acc, c.W1, c.b1, c.W2, c.b2, c.WF, c.bF)
      if      (c.K == 64)            LAUNCH_CONV(64, true);
      else if (c.K == 32 && c.three) LAUNCH_CONV(32, true);
      else if (c.K == 32)            LAUNCH_CONV(32, false);
      else if (c.three)              LAUNCH_CONV(16, true);
      else                           LAUNCH_CONV(16, false);
#undef LAUNCH_CONV
      node_update_kernel<<<(NN * WID + TB - 1) / TB, TB, 0, stream>>>(
          h, acc, invc + (long)c.seg * NN);
    }
  }

  fc_out1_kernel<<<(NP0 * 64 + TB - 1) / TB, TB, 0, stream>>>(h, fo1_W, fo1_b, h2);
  fc_out2_kernel<<<(NP0 + TB - 1) / TB, TB, 0, stream>>>(h2, fo2_W, fo2_b, (float*)d_out);
}